// Temporal_GCN_21234318311493
// MI455X (gfx1250) — compile-verified
//
#include <hip/hip_runtime.h>
#include <math.h>

namespace {
constexpr int NB = 32;    // batch
constexpr int NC = 64;    // in channels
constexpr int NT = 256;   // time
constexpr int NV = 25;    // vertices
constexpr int NRC = 16;   // rel channels (C/4)
constexpr int NBC = 16;   // branch channels (64/H)
constexpr int NK = 400;   // NRC * NV (K of the At GEMM)
constexpr int NTV = NT * NV;
constexpr float BN_EPS = 1e-5f;
}

typedef __attribute__((ext_vector_type(2))) float v2f;
typedef __attribute__((ext_vector_type(8))) float v8f;

// ---------------------------------------------------------------------------
// Stage 1: per-head 1x1 convs -> y1t / y2t laid out K-major: y[b][t][o*NV+v]
// ---------------------------------------------------------------------------
__global__ void __launch_bounds__(256)
k_blk(const float* __restrict__ x,
      const float* __restrict__ w1, const float* __restrict__ b1,
      const float* __restrict__ w2, const float* __restrict__ b2,
      float* __restrict__ y1t, float* __restrict__ y2t) {
  __shared__ float sw1[NRC * NC];
  __shared__ float sw2[NRC * NC];
  for (int i = threadIdx.x; i < NRC * NC; i += blockDim.x) {
    sw1[i] = w1[i];
    sw2[i] = w2[i];
  }
  __syncthreads();
  int gid = blockIdx.x * blockDim.x + threadIdx.x;
  if (gid >= NB * NT * NV) return;
  int v = gid % NV;
  int t = (gid / NV) % NT;
  int b = gid / (NT * NV);
  const float* xp = x + (b * NC) * NTV + t * NV + v;
  float a1[NRC], a2[NRC];
#pragma unroll
  for (int o = 0; o < NRC; ++o) { a1[o] = 0.f; a2[o] = 0.f; }
  for (int c = 0; c < NC; ++c) {
    float xv = xp[c * NTV];
#pragma unroll
    for (int o = 0; o < NRC; ++o) {
      a1[o] += sw1[o * NC + c] * xv;
      a2[o] += sw2[o * NC + c] * xv;
    }
  }
  float* p1 = y1t + (b * NT + t) * NK + v;
  float* p2 = y2t + (b * NT + t) * NK + v;
#pragma unroll
  for (int o = 0; o < NRC; ++o) {
    p1[o * NV] = a1[o] + b1[o];
    p2[o * NV] = a2[o] + b2[o];
  }
}

// ---------------------------------------------------------------------------
// Stage 2: At[b,t,g] = sum_k y1t[b,t,k] * y2t[b,g,k]   (WMMA f32 16x16x4)
// one wave per 32x32 output super-tile (2x2 register blocking), K = 400
// ---------------------------------------------------------------------------
__global__ void __launch_bounds__(128)
k_at(const float* __restrict__ y1t, const float* __restrict__ y2t,
     float* __restrict__ At) {
  int wave = (blockIdx.x * blockDim.x + threadIdx.x) >> 5;
  int lane = threadIdx.x & 31;
  if (wave >= NB * 8 * 8) return;
  int tn2 = wave & 7;            // g super-tile (32 wide)
  int tm2 = (wave >> 3) & 7;     // t super-tile (32 tall)
  int b   = wave >> 6;

  int ml = lane & 15;            // M row (A) / N col (B)
  int kb = (lane >> 4) << 1;     // lanes 0-15 -> K{0,1}, lanes 16-31 -> K{2,3}

  const float* ap[2];
  const float* bp[2];
#pragma unroll
  for (int i = 0; i < 2; ++i) {
    ap[i] = y1t + (b * NT + tm2 * 32 + i * 16 + ml) * NK + kb;
    bp[i] = y2t + (b * NT + tn2 * 32 + i * 16 + ml) * NK + kb;
  }

  v8f acc[2][2];
#pragma unroll
  for (int mi = 0; mi < 2; ++mi)
#pragma unroll
    for (int ni = 0; ni < 2; ++ni) acc[mi][ni] = {};

  for (int k0 = 0; k0 < NK; k0 += 4) {
    v2f af[2], bf[2];
#pragma unroll
    for (int i = 0; i < 2; ++i) {
      af[i] = *(const v2f*)(ap[i] + k0);
      bf[i] = *(const v2f*)(bp[i] + k0);
    }
#pragma unroll
    for (int mi = 0; mi < 2; ++mi)
#pragma unroll
      for (int ni = 0; ni < 2; ++ni)
        acc[mi][ni] = __builtin_amdgcn_wmma_f32_16x16x4_f32(
            false, af[mi], false, bf[ni], (short)0, acc[mi][ni], false, false);
  }

  // D layout: N = lane&15 ; M = r + 8*(lane>=16)
  int hi = (lane >> 4) * 8;
#pragma unroll
  for (int mi = 0; mi < 2; ++mi) {
#pragma unroll
    for (int ni = 0; ni < 2; ++ni) {
      float* dp = At + (b * NT + tm2 * 32 + mi * 16 + hi) * NT +
                  tn2 * 32 + ni * 16 + ml;
#pragma unroll
      for (int r = 0; r < 8; ++r) dp[r * NT] = acc[mi][ni][r];
    }
  }
}

// ---------------------------------------------------------------------------
// Stage 3: softmax over t (axis=1) for each (b,g), then += masked WA.
// Optionally mirrors result into the temporal_A output region (head 3).
// ---------------------------------------------------------------------------
__global__ void __launch_bounds__(256)
k_softmax(float* __restrict__ At, const float* __restrict__ WA,
          float* __restrict__ outTA, int writeOut) {
  int gid = blockIdx.x * blockDim.x + threadIdx.x;
  if (gid >= NB * NT) return;
  int g = gid % NT;
  int b = gid / NT;
  float* col = At + b * NT * NT + g;   // stride NT over t
  float mx = -INFINITY;
  for (int t = 0; t < NT; ++t) mx = fmaxf(mx, col[t * NT]);
  float s = 0.f;
  for (int t = 0; t < NT; ++t) {
    float e = __expf(col[t * NT] - mx);
    s += e;
    col[t * NT] = e;
  }
  float inv = 1.f / s;
  for (int t = 0; t < NT; ++t) {
    float aw = (g <= t + 1) ? WA[t * NT + g] : 0.f;
    float val = col[t * NT] * inv + aw;
    col[t * NT] = val;
    if (writeOut) outTA[b * NT * NT + t * NT + g] = val;
  }
}

// ---------------------------------------------------------------------------
// Stage 4: z[b,c,g,v] = sum_t A[b,t,g] * x[b,c,t,v]   (WMMA f32 16x16x4)
// one wave per 64x64 output super-tile (4x4 register blocking).
// M = (c,v) = 1600 rows (25 super-tiles), N = g = 256 (4 super-tiles), K = 256
// ---------------------------------------------------------------------------
__global__ void __launch_bounds__(128)
k_z(const float* __restrict__ x, const float* __restrict__ At,
    float* __restrict__ z) {
  int wave = (blockIdx.x * blockDim.x + threadIdx.x) >> 5;
  int lane = threadIdx.x & 31;
  if (wave >= NB * 25 * 4) return;
  int tn4 = wave & 3;            // g super-tile (64 wide)
  int tm4 = (wave >> 2) % 25;    // (c,v) super-tile (64 tall)
  int b   = wave / 100;

  int ml = lane & 15;
  int kb = (lane >> 4) << 1;

  const float* ap[4];
#pragma unroll
  for (int mi = 0; mi < 4; ++mi) {
    int m = tm4 * 64 + mi * 16 + ml;
    int ca = m / NV, va = m % NV;
    ap[mi] = x + (b * NC + ca) * NTV + kb * NV + va;  // + t*NV per k-step
  }
  const float* bp[4];
#pragma unroll
  for (int ni = 0; ni < 4; ++ni)
    bp[ni] = At + (b * NT + kb) * NT + tn4 * 64 + ni * 16 + ml;  // + t*NT per step

  v8f acc[4][4];
#pragma unroll
  for (int mi = 0; mi < 4; ++mi)
#pragma unroll
    for (int ni = 0; ni < 4; ++ni) acc[mi][ni] = {};

  for (int k0 = 0; k0 < NT; k0 += 4) {
    // prefetch the strided x stream two k-steps (8 t) ahead
    __builtin_prefetch(ap[0] + (k0 + 8) * NV, 0, 0);
    v2f af[4], bf[4];
#pragma unroll
    for (int mi = 0; mi < 4; ++mi) {
      af[mi].x = ap[mi][k0 * NV];
      af[mi].y = ap[mi][(k0 + 1) * NV];
    }
#pragma unroll
    for (int ni = 0; ni < 4; ++ni) {
      bf[ni].x = bp[ni][k0 * NT];
      bf[ni].y = bp[ni][(k0 + 1) * NT];
    }
#pragma unroll
    for (int mi = 0; mi < 4; ++mi)
#pragma unroll
      for (int ni = 0; ni < 4; ++ni)
        acc[mi][ni] = __builtin_amdgcn_wmma_f32_16x16x4_f32(
            false, af[mi], false, bf[ni], (short)0, acc[mi][ni], false, false);
  }

  int n = lane & 15;
  int hi = (lane >> 4) * 8;
#pragma unroll
  for (int mi = 0; mi < 4; ++mi) {
#pragma unroll
    for (int r = 0; r < 8; ++r) {
      int mr = tm4 * 64 + mi * 16 + r + hi;
      int c2 = mr / NV, v2 = mr % NV;
      float* zp = z + ((b * NC + c2) * NT + tn4 * 64 + n) * NV + v2;
#pragma unroll
      for (int ni = 0; ni < 4; ++ni) zp[ni * 16 * NV] = acc[mi][ni][r];
    }
  }
}

// ---------------------------------------------------------------------------
// Stage 5: h = relu(BN1(pw_w @ z + pw_b)); BN folded into scale/shift.
// outC/coff parameterize destination (workspace h, or d_out block for head 3)
// ---------------------------------------------------------------------------
__global__ void __launch_bounds__(256)
k_h(const float* __restrict__ z,
    const float* __restrict__ pw_w, const float* __restrict__ pw_b,
    const float* __restrict__ bg, const float* __restrict__ bb,
    const float* __restrict__ bm, const float* __restrict__ bv,
    float* __restrict__ out, int outC, int coff) {
  __shared__ float sw[NBC * NC];
  __shared__ float sscale[NBC];
  __shared__ float sshift[NBC];
  for (int i = threadIdx.x; i < NBC * NC; i += blockDim.x) sw[i] = pw_w[i];
  if (threadIdx.x < NBC) {
    int o = threadIdx.x;
    float sc = bg[o] * rsqrtf(bv[o] + BN_EPS);
    sscale[o] = sc;
    sshift[o] = (pw_b[o] - bm[o]) * sc + bb[o];
  }
  __syncthreads();
  int gid = blockIdx.x * blockDim.x + threadIdx.x;
  if (gid >= NB * NT * NV) return;
  int v = gid % NV;
  int t = (gid / NV) % NT;
  int b = gid / (NT * NV);
  const float* zp = z + (b * NC) * NTV + t * NV + v;
  float acc[NBC];
#pragma unroll
  for (int o = 0; o < NBC; ++o) acc[o] = 0.f;
  for (int c = 0; c < NC; ++c) {
    float zv = zp[c * NTV];
#pragma unroll
    for (int o = 0; o < NBC; ++o) acc[o] += sw[o * NC + c] * zv;
  }
#pragma unroll
  for (int o = 0; o < NBC; ++o) {
    float hv = fmaxf(acc[o] * sscale[o] + sshift[o], 0.f);
    out[(b * outC + coff + o) * NTV + t * NV + v] = hv;
  }
}

// ---------------------------------------------------------------------------
// Branch i<2: dilated temporal conv (KS=3, pad=dil) + BN -> d_out block
// ---------------------------------------------------------------------------
__global__ void __launch_bounds__(256)
k_tconv(const float* __restrict__ h, const float* __restrict__ w,
        const float* __restrict__ cb,
        const float* __restrict__ bg, const float* __restrict__ bb,
        const float* __restrict__ bm, const float* __restrict__ bv,
        float* __restrict__ out, int dil, int coff) {
  int gid = blockIdx.x * blockDim.x + threadIdx.x;
  if (gid >= NB * NBC * NT * NV) return;
  int v = gid % NV;
  int t = (gid / NV) % NT;
  int o = (gid / (NT * NV)) % NBC;
  int b = gid / (NBC * NT * NV);
  float acc = 0.f;
#pragma unroll
  for (int ks = 0; ks < 3; ++ks) {
    int tt = t + dil * (ks - 1);
    if (tt < 0 || tt >= NT) continue;
    const float* hp = h + (b * NBC) * NTV + tt * NV + v;
    const float* wp = w + o * NBC * 3 + ks;
#pragma unroll
    for (int ic = 0; ic < NBC; ++ic) acc += wp[ic * 3] * hp[ic * NTV];
  }
  acc += cb[o];
  float sc = bg[o] * rsqrtf(bv[o] + BN_EPS);
  out[(b * NC + coff + o) * NTV + t * NV + v] = (acc - bm[o]) * sc + bb[o];
}

// ---------------------------------------------------------------------------
// Branch i==2: maxpool3 over t (pad 1) + BN -> d_out block
// ---------------------------------------------------------------------------
__global__ void __launch_bounds__(256)
k_maxpool(const float* __restrict__ h,
          const float* __restrict__ bg, const float* __restrict__ bb,
          const float* __restrict__ bm, const float* __restrict__ bv,
          float* __restrict__ out, int coff) {
  int gid = blockIdx.x * blockDim.x + threadIdx.x;
  if (gid >= NB * NBC * NT * NV) return;
  int v = gid % NV;
  int t = (gid / NV) % NT;
  int o = (gid / (NT * NV)) % NBC;
  int b = gid / (NBC * NT * NV);
  const float* hp = h + (b * NBC + o) * NTV + v;
  float m = -INFINITY;
  if (t > 0) m = hp[(t - 1) * NV];
  m = fmaxf(m, hp[t * NV]);
  if (t < NT - 1) m = fmaxf(m, hp[(t + 1) * NV]);
  float sc = bg[o] * rsqrtf(bv[o] + BN_EPS);
  out[(b * NC + coff + o) * NTV + t * NV + v] = (m - bm[o]) * sc + bb[o];
}

// ---------------------------------------------------------------------------
// AW output: AW[t,g] = (g <= t+1) ? WA[t,g] : 0
// ---------------------------------------------------------------------------
__global__ void __launch_bounds__(256)
k_aw(const float* __restrict__ WA, float* __restrict__ outAW) {
  int gid = blockIdx.x * blockDim.x + threadIdx.x;
  if (gid >= NT * NT) return;
  int g = gid % NT;
  int t = gid / NT;
  outAW[gid] = (g <= t + 1) ? WA[gid] : 0.f;
}

// ---------------------------------------------------------------------------
extern "C" void kernel_launch(void* const* d_in, const int* in_sizes, int n_in,
                              void* d_out, int out_size, void* d_ws, size_t ws_size,
                              hipStream_t stream) {
  (void)in_sizes; (void)n_in; (void)out_size; (void)ws_size;
  const float* x      = (const float*)d_in[0];
  const float* WA     = (const float*)d_in[1];
  const float* blk_w1 = (const float*)d_in[2];
  const float* blk_b1 = (const float*)d_in[3];
  const float* blk_w2 = (const float*)d_in[4];
  const float* blk_b2 = (const float*)d_in[5];
  const float* pw_w   = (const float*)d_in[6];
  const float* pw_b   = (const float*)d_in[7];
  const float* bn1_g  = (const float*)d_in[8];
  const float* bn1_b  = (const float*)d_in[9];
  const float* bn1_m  = (const float*)d_in[10];
  const float* bn1_v  = (const float*)d_in[11];
  const float* tc_w   = (const float*)d_in[12];
  const float* tc_b   = (const float*)d_in[13];
  const float* tcbn_g = (const float*)d_in[14];
  const float* tcbn_b = (const float*)d_in[15];
  const float* tcbn_m = (const float*)d_in[16];
  const float* tcbn_v = (const float*)d_in[17];
  const float* mpbn_g = (const float*)d_in[18];
  const float* mpbn_b = (const float*)d_in[19];
  const float* mpbn_m = (const float*)d_in[20];
  const float* mpbn_v = (const float*)d_in[21];

  float* out0  = (float*)d_out;                           // (B,64,T,V)
  float* outTA = out0 + (size_t)NB * NC * NT * NV;        // (B,T,T)
  float* outAW = outTA + (size_t)NB * NT * NT;            // (T,T)

  // workspace carve-up
  float* ws   = (float*)d_ws;
  float* y1t  = ws;                                       // B*T*400
  float* y2t  = y1t + (size_t)NB * NT * NK;               // B*T*400
  float* At   = y2t + (size_t)NB * NT * NK;               // B*T*T
  float* zbuf = At + (size_t)NB * NT * NT;                // B*C*T*V
  float* hbuf = zbuf + (size_t)NB * NC * NT * NV;         // B*16*T*V

  const int posN   = NB * NT * NV;              // 204800
  const int posBlk = (posN + 255) / 256;
  const int atWaves = NB * 8 * 8;               // 2048 waves (32x32 tiles)
  const int atBlk   = (atWaves * 32 + 127) / 128;
  const int zWaves  = NB * 25 * 4;              // 3200 waves (64x64 tiles)
  const int zBlk    = (zWaves * 32 + 127) / 128;
  const int smN   = NB * NT;                    // 8192
  const int smBlk = (smN + 255) / 256;
  const int brN   = NB * NBC * NT * NV;         // 3276800
  const int brBlk = (brN + 255) / 256;

  k_aw<<<(NT * NT + 255) / 256, 256, 0, stream>>>(WA, outAW);

  for (int head = 0; head < 4; ++head) {
    k_blk<<<posBlk, 256, 0, stream>>>(
        x,
        blk_w1 + head * NRC * NC, blk_b1 + head * NRC,
        blk_w2 + head * NRC * NC, blk_b2 + head * NRC,
        y1t, y2t);

    k_at<<<atBlk, 128, 0, stream>>>(y1t, y2t, At);

    k_softmax<<<smBlk, 256, 0, stream>>>(At, WA, outTA, head == 3 ? 1 : 0);

    k_z<<<zBlk, 128, 0, stream>>>(x, At, zbuf);

    if (head == 3) {
      // plain branch: write h straight into the output channel block
      k_h<<<posBlk, 256, 0, stream>>>(
          zbuf, pw_w + head * NBC * NC, pw_b + head * NBC,
          bn1_g + head * NBC, bn1_b + head * NBC,
          bn1_m + head * NBC, bn1_v + head * NBC,
          out0, NC, head * NBC);
    } else {
      k_h<<<posBlk, 256, 0, stream>>>(
          zbuf, pw_w + head * NBC * NC, pw_b + head * NBC,
          bn1_g + head * NBC, bn1_b + head * NBC,
          bn1_m + head * NBC, bn1_v + head * NBC,
          hbuf, NBC, 0);
      if (head < 2) {
        int dil = (head == 0) ? 1 : 2;
        k_tconv<<<brBlk, 256, 0, stream>>>(
            hbuf, tc_w + head * NBC * NBC * 3, tc_b + head * NBC,
            tcbn_g + head * NBC, tcbn_b + head * NBC,
            tcbn_m + head * NBC, tcbn_v + head * NBC,
            out0, dil, head * NBC);
      } else {  // head == 2
        k_maxpool<<<brBlk, 256, 0, stream>>>(
            hbuf, mpbn_g, mpbn_b, mpbn_m, mpbn_v, out0, head * NBC);
      }
    }
  }
}